// SineOscillator_5918464934242
// MI455X (gfx1250) — compile-verified
//
#include <hip/hip_runtime.h>
#include <stdint.h>

// ---------------------------------------------------------------------------
// SineOscillator for MI455X (gfx1250, wave32)
//
// out[b, j] = sin(2*pi * frac(cumsum_j(f0_up/24000))) * (f0_up >= 20) + 0.05*noise
//
// f0_up = linear-upsample x480 of f0[32][2000]. Within a 480-sample frame f0_up
// is two linear ramps, so the per-frame phase step and the within-frame
// inclusive prefix have exact closed forms -> no per-sample serial scan:
//   frame step     = (60*fL + 360*fC + 60*fR)/24000           (edges: clamp)
//   prefix, t<240 : [fL*(t+1)(239.5-t/2) + fC*(t+1)(t/2+240.5)] / (480*24000)
//   prefix, t>=240: (60fL+180fC)/24000
//                 + [fC*(m+1)(479.5-m/2) + fR*(m+1)(m/2+0.5)] / (480*24000)
// Both halves share the form fA*(c1-m) + fB*(m+c2) -> branchless via selects.
//
// Bandwidth bound: ~246 MB traffic -> ~10.6 us at 23.3 TB/s. No GEMM content,
// so no WMMA; gfx1250 paths: async global->LDS DMA (ASYNCcnt), wave32 shuffle
// scan, v_sin_f32 (computes sin(2*pi*x), saving the 2*pi multiply), NT stores
// (streaming set > 192MB L2).
// ---------------------------------------------------------------------------

typedef float __attribute__((ext_vector_type(4))) f32x4;

#define BATCH   32
#define LFR     2000
#define FRAME   480
#define TLEN    (LFR * FRAME)          /* 960000 samples per row            */
#define GROUPS_PER_ROW (TLEN / 4)      /* 240000 float4 groups per row      */
#define MAIN_BLOCK 256
#define MAIN_GRID  ((BATCH * GROUPS_PER_ROW) / MAIN_BLOCK) /* 30000, exact  */

static_assert((BATCH * GROUPS_PER_ROW) % MAIN_BLOCK == 0, "exact grid");
static_assert(FRAME % 4 == 0 && 240 % 4 == 0, "float4 groups never straddle");

// ---------------------------------------------------------------------------
// Kernel A: per-row frame-level phase scan. One wave32 per batch row.
// Lane j partial-sums frames [63j, 63j+63) in f64, wave shuffle-scan, then
// writes the *fractional* exclusive phase offset of every frame (frac keeps
// magnitudes < 1, so the f32 offsets lose no phase precision).
// ---------------------------------------------------------------------------
__global__ __launch_bounds__(32) void sine_frame_scan(const float* __restrict__ f0,
                                                      float* __restrict__ offs) {
  const int b    = blockIdx.x;     // 0..31
  const int lane = threadIdx.x;    // 0..31
  const float* f = f0 + b * LFR;

  const int k0 = lane * 63;
  const int k1 = (k0 + 63 < LFR) ? (k0 + 63) : LFR;

  double s = 0.0;
  for (int k = k0; k < k1; ++k) {
    float fC = f[k];
    float fL = (k > 0)       ? f[k - 1] : fC;   // clamp reproduces edge frames
    float fR = (k < LFR - 1) ? f[k + 1] : fC;
    s += 60.0 * (double)fL + 360.0 * (double)fC + 60.0 * (double)fR;
  }

  // inclusive wave32 scan of lane partial sums
  double inc = s;
  #pragma unroll
  for (int d = 1; d < 32; d <<= 1) {
    double up = __shfl_up(inc, d, 32);
    if (lane >= d) inc += up;
  }

  double acc = (inc - s) * (1.0 / 24000.0);     // exclusive prefix in cycles
  for (int k = k0; k < k1; ++k) {
    offs[b * LFR + k] = (float)(acc - floor(acc));
    float fC = f[k];
    float fL = (k > 0)       ? f[k - 1] : fC;
    float fR = (k < LFR - 1) ? f[k + 1] : fC;
    acc += (60.0 * (double)fL + 360.0 * (double)fC + 60.0 * (double)fR) * (1.0 / 24000.0);
  }
}

// ---------------------------------------------------------------------------
// Kernel B: main pointwise kernel, one thread per float4 output group.
// Noise is DMA'd to LDS with global_load_async_to_lds_b128 (overlaps the
// phase math); output streamed with non-temporal stores.
// ---------------------------------------------------------------------------
__global__ __launch_bounds__(MAIN_BLOCK) void sine_osc_main(const float* __restrict__ f0,
                                                            const float* __restrict__ noise,
                                                            const float* __restrict__ offs,
                                                            float* __restrict__ out) {
  __shared__ f32x4 sm[MAIN_BLOCK];

  const int tid = threadIdx.x;
  const int gid = blockIdx.x * MAIN_BLOCK + tid;
  const int b   = gid / GROUPS_PER_ROW;
  const int g   = gid - b * GROUPS_PER_ROW;
  const int s0  = g * 4;                // first sample of this group in row
  const int k   = s0 / FRAME;           // frame index
  const int t0  = s0 - k * FRAME;       // 0..476, multiple of 4

  // --- launch async DMA: 16B/lane of noise -> LDS (GVS: SGPR base + v-offset)
  {
    const unsigned voff = (unsigned)gid * 16u;         // b*TLEN+s0 == gid*4 elems
    const unsigned ldsa = (unsigned)(uintptr_t)(&sm[tid]); // low 32b = LDS offset
    asm volatile("global_load_async_to_lds_b128 %0, %1, %2 offset:0"
                 :: "v"(ldsa), "v"(voff), "s"(noise)
                 : "memory");
  }

  // --- phase math (overlaps the DMA)
  const float* f = f0 + b * LFR;        // 8KB/row, stays cache-resident
  const float fC = f[k];
  const float fL = (k > 0)       ? f[k - 1] : fC;
  const float fR = (k < LFR - 1) ? f[k + 1] : fC;
  const float off = offs[b * LFR + k];

  const float inv480  = 1.0f / 480.0f;
  const float invPh   = 1.0f / (480.0f * 24000.0f);
  const float seg1tot = (60.0f * fL + 180.0f * fC) * (1.0f / 24000.0f);

  // Branchless half-frame coefficients (uniform per 4-sample group).
  const bool h2   = (t0 >= 240);        // second linear ramp?
  const float fA  = h2 ? fC : fL;
  const float fB  = h2 ? fR : fC;
  const float c1  = h2 ? 479.5f : 239.5f;
  const float c2  = h2 ? 0.5f   : 240.5f;
  const float bse = h2 ? seg1tot : 0.0f;
  const float m0  = (float)(h2 ? (t0 - 240) : t0);

  float sv[4], uv[4];
  #pragma unroll
  for (int i = 0; i < 4; ++i) {
    const float mf  = m0 + (float)i;
    const float mp1 = mf + 1.0f;
    const float val   = (fA * (c1 - mf) + fB * (mf + c2)) * inv480;     // f0_up
    const float local = bse + (fA * (mp1 * (c1 - 0.5f * mf)) +
                               fB * (mp1 * (0.5f * mf + c2))) * invPh;  // incl. prefix
    const float phase = off + local;            // >= 0 always
    const float fr    = phase - floorf(phase);  // frac -> revolutions
    sv[i] = __builtin_amdgcn_sinf(fr);          // v_sin_f32: sin(2*pi*x)
    uv[i] = (val >= 20.0f) ? 1.0f : 0.0f;       // voiced mask
  }

  // --- consume DMA'd noise
  asm volatile("s_wait_asynccnt 0" ::: "memory");
  const f32x4 nz = sm[tid];

  f32x4 o;
  o.x = sv[0] * uv[0] + nz.x * 0.05f;
  o.y = sv[1] * uv[1] + nz.y * 0.05f;
  o.z = sv[2] * uv[2] + nz.z * 0.05f;
  o.w = sv[3] * uv[3] + nz.w * 0.05f;

  __builtin_nontemporal_store(o, (f32x4*)(out + (size_t)gid * 4));
}

// ---------------------------------------------------------------------------
extern "C" void kernel_launch(void* const* d_in, const int* in_sizes, int n_in,
                              void* d_out, int out_size, void* d_ws, size_t ws_size,
                              hipStream_t stream) {
  const float* f0    = (const float*)d_in[0];   // [32,1,2000]
  const float* noise = (const float*)d_in[1];   // [32,1,960000]
  float*       out   = (float*)d_out;           // [32,1,960000]
  float*       offs  = (float*)d_ws;            // 32*2000 floats = 256 KB scratch

  sine_frame_scan<<<BATCH, 32, 0, stream>>>(f0, offs);
  sine_osc_main<<<MAIN_GRID, MAIN_BLOCK, 0, stream>>>(f0, noise, offs, out);
}